// QGCNLayer_v2_88905823027435
// MI455X (gfx1250) — compile-verified
//
#include <hip/hip_runtime.h>
#include <hip/hip_bf16.h>
#include <math.h>

#define N_NODES 100000
#define N_EDGES 1600000
#define BN_EPS 1e-5f

typedef __attribute__((ext_vector_type(2))) float v2f;
typedef __attribute__((ext_vector_type(8))) float v8f;

// ---------------- kernel 0: build permuted hamilton [64][64] ----------------
__global__ void build_hamilton_kernel(const float* __restrict__ weight,
                                      float* __restrict__ H) {
  __shared__ int inv[64];
  int tid = threadIdx.x;
  if (tid < 64) {
    int p = (tid * 4) % 64 + (tid * 4) / 64;
    inv[p] = tid;
  }
  __syncthreads();
  const int  comp[4][4] = {{0,1,2,3},{1,0,3,2},{2,3,0,1},{3,2,1,0}};
  const float sgn[4][4] = {{1.f,-1.f,-1.f,-1.f},
                           {1.f, 1.f,-1.f, 1.f},
                           {1.f, 1.f, 1.f,-1.f},
                           {1.f,-1.f, 1.f, 1.f}};
  for (int idx = tid; idx < 64 * 64; idx += blockDim.x) {
    int ii = idx >> 6, jj = idx & 63;
    int ri = inv[ii], cj = inv[jj];
    int q = ri >> 4, a = ri & 15;
    int c = cj >> 4, b = cj & 15;
    H[idx] = sgn[q][c] * weight[a * 64 + comp[q][c] * 16 + b];
  }
}

// ---------------- kernel 1: zero output + reduction scratch ----------------
__global__ void zero_out_kernel(float* __restrict__ out,
                                float* __restrict__ small, long n4) {
  long stride = (long)gridDim.x * blockDim.x;
  for (long k = (long)blockIdx.x * blockDim.x + threadIdx.x; k < n4; k += stride)
    ((float4*)out)[k] = make_float4(0.f, 0.f, 0.f, 0.f);
  if (blockIdx.x == 0 && threadIdx.x < 256) small[threadIdx.x] = 0.f;
}

// ---------------- kernel 2: support = x @ H via V_WMMA_F32_16X16X4_F32 ------
__global__ void __launch_bounds__(128)
qgcn_gemm_wmma(const float* __restrict__ x, const float* __restrict__ H,
               float* __restrict__ support) {
  __shared__ float Hs[64 * 64];
  int tid = threadIdx.x;
  // Stage 16KB hamilton into LDS: 1024 float4 / 128 threads = 8 each.
  const float4* Hg4 = (const float4*)H;
  float4* Hs4 = (float4*)Hs;
#pragma unroll
  for (int i = 0; i < 8; ++i) Hs4[tid + i * 128] = Hg4[tid + i * 128];
  __syncthreads();

  int wave = tid >> 5;            // 0..3 -> 16-column output tile
  int lane = tid & 31;
  int half = lane >> 4;           // ISA A layout: lanes 0-15 hold K=0,1; 16-31 hold K=2,3
  int m = lane & 15;
  int colbase = wave * 16;
  long rowbase = (long)blockIdx.x * 16;
  const float* xrow = x + (rowbase + m) * 64;

  v8f acc = {};
#pragma unroll
  for (int kt = 0; kt < 16; ++kt) {
    int k0 = kt * 4 + 2 * half;
    v2f a = *(const v2f*)(xrow + k0);          // A: 16x4 fp32 fragment
    v2f b;
    b.x = Hs[(k0    ) * 64 + colbase + m];     // B: 4x16 fp32 fragment
    b.y = Hs[(k0 + 1) * 64 + colbase + m];
    acc = __builtin_amdgcn_wmma_f32_16x16x4_f32(
        /*neg_a=*/false, a, /*neg_b=*/false, b,
        /*c_mod=*/(short)0, acc, /*reuse_a=*/false, /*reuse_b=*/false);
  }
  // C/D layout: VGPR r holds M=r (lanes 0-15) and M=r+8 (lanes 16-31), N=lane%16.
#pragma unroll
  for (int r = 0; r < 8; ++r) {
    long row = rowbase + r + 8 * half;
    support[row * 64 + colbase + m] = acc[r];
  }
}

// ---------------- kernel 3: COO SpMM with f32 atomics (L2-resident) ---------
__global__ void __launch_bounds__(256)
spmm_kernel(const int* __restrict__ erow, const int* __restrict__ ecol,
            const float* __restrict__ eval, const float* __restrict__ support,
            float* __restrict__ out) {
  int lane = threadIdx.x & 31;
  long wave = ((long)blockIdx.x * blockDim.x + threadIdx.x) >> 5;
  long nwaves = ((long)gridDim.x * blockDim.x) >> 5;
  for (long e = wave; e < (long)N_EDGES; e += nwaves) {
    int col = ecol[e];
    int row = erow[e];
    float val = eval[e];
    float2 s = *(const float2*)(support + (long)col * 64 + lane * 2);
    float* dst = out + (long)row * 64 + lane * 2;
    atomicAdd(dst,     val * s.x);
    atomicAdd(dst + 1, val * s.y);
  }
}

// ---------------- kernel 4: per-feature partial sums for BN -----------------
__global__ void __launch_bounds__(256)
bn_partial_kernel(const float* __restrict__ out, float* __restrict__ sums,
                  float* __restrict__ sumsq) {
  __shared__ float ls[256], ls2[256];
  int f = threadIdx.x & 63;
  int sub = threadIdx.x >> 6;   // 0..3
  long rbase = (long)blockIdx.x * 512;
  long rend = rbase + 512; if (rend > N_NODES) rend = N_NODES;
  float s = 0.f, s2 = 0.f;
  for (long r = rbase + sub; r < rend; r += 4) {
    float v = out[r * 64 + f];
    s += v; s2 += v * v;
  }
  ls[threadIdx.x] = s; ls2[threadIdx.x] = s2;
  __syncthreads();
  if (sub == 0) {
    s  = ls[f]  + ls[f + 64]  + ls[f + 128]  + ls[f + 192];
    s2 = ls2[f] + ls2[f + 64] + ls2[f + 128] + ls2[f + 192];
    atomicAdd(&sums[f], s);
    atomicAdd(&sumsq[f], s2);
  }
}

// ---------------- kernel 5: BN coefficients ---------------------------------
__global__ void bn_finalize_kernel(const float* __restrict__ sums,
                                   const float* __restrict__ sumsq,
                                   const float* __restrict__ gamma,
                                   const float* __restrict__ beta,
                                   float* __restrict__ coefA,
                                   float* __restrict__ coefB) {
  int f = threadIdx.x;
  if (f < 64) {
    float mean = sums[f] * (1.0f / (float)N_NODES);
    float var = sumsq[f] * (1.0f / (float)N_NODES) - mean * mean;
    float rstd = rsqrtf(var + BN_EPS);
    float a = gamma[f] * rstd;
    coefA[f] = a;
    coefB[f] = beta[f] - mean * a;
  }
}

// ---------------- kernel 6: apply BN + tanh in place ------------------------
__global__ void bn_apply_kernel(float* __restrict__ out,
                                const float* __restrict__ coefA,
                                const float* __restrict__ coefB) {
  long tot = (long)N_NODES * 64;
  long stride = (long)gridDim.x * blockDim.x;
  for (long i = (long)blockIdx.x * blockDim.x + threadIdx.x; i < tot; i += stride) {
    int f = (int)(i & 63);
    out[i] = tanhf(coefA[f] * out[i] + coefB[f]);
  }
}

extern "C" void kernel_launch(void* const* d_in, const int* in_sizes, int n_in,
                              void* d_out, int out_size, void* d_ws, size_t ws_size,
                              hipStream_t stream) {
  (void)in_sizes; (void)n_in; (void)out_size; (void)ws_size;
  const float* x      = (const float*)d_in[0];
  const int*   erow   = (const int*)d_in[1];
  const int*   ecol   = (const int*)d_in[2];
  const float* eval   = (const float*)d_in[3];
  const float* weight = (const float*)d_in[4];
  const float* gamma  = (const float*)d_in[5];
  const float* beta   = (const float*)d_in[6];
  float* out = (float*)d_out;

  char* ws = (char*)d_ws;
  float* H       = (float*)ws;                                   // 4096 floats
  float* support = (float*)(ws + 16384);                         // N*64 floats
  float* small   = (float*)(ws + 16384 + (size_t)N_NODES * 64 * 4);
  float* sums  = small;
  float* sumsq = small + 64;
  float* coefA = small + 128;
  float* coefB = small + 192;

  build_hamilton_kernel<<<1, 256, 0, stream>>>(weight, H);
  zero_out_kernel<<<2048, 256, 0, stream>>>(out, small, (long)N_NODES * 64 / 4);
  qgcn_gemm_wmma<<<N_NODES / 16, 128, 0, stream>>>(x, H, support);
  spmm_kernel<<<4096, 256, 0, stream>>>(erow, ecol, eval, support, out);
  bn_partial_kernel<<<(N_NODES + 511) / 512, 256, 0, stream>>>(out, sums, sumsq);
  bn_finalize_kernel<<<1, 64, 0, stream>>>(sums, sumsq, gamma, beta, coefA, coefB);
  bn_apply_kernel<<<2048, 256, 0, stream>>>(out, coefA, coefB);
}